// PointPillarScatter_52536039964810
// MI455X (gfx1250) — compile-verified
//
#include <hip/hip_runtime.h>

// PointPillarScatter for MI455X (gfx1250, wave32).
// Gather formulation: build cell->pillar inverse map in workspace, then stream
// the output canvas with coalesced nontemporal stores while gathering pillar
// features (L2-resident) with float4 loads. Map tiles are staged into LDS via
// the CDNA5 async global->LDS path (ASYNCcnt / s_wait_asynccnt).

#define NXC 432
#define NYC 496
#define NCELL (NXC * NYC) // 214272, divisible by 64

// ---- Pass 1a: init inverse map to -1 ------------------------------------
__global__ void pps_init_map(int* __restrict__ map, int n) {
    int i = blockIdx.x * blockDim.x + threadIdx.x;
    if (i < n) map[i] = -1;
}

// ---- Pass 1b: scatter pillar index into map ------------------------------
__global__ void pps_build_map(const long long* __restrict__ coords,
                              int* __restrict__ map, int B, int P) {
    int i = blockIdx.x * blockDim.x + threadIdx.x;
    if (i >= B * P) return;
    int b = i / P;
    int p = i - b * P;
    const long long* c4 = coords + (size_t)i * 4; // [b, z, y, x]
    int idx = (int)c4[2] * NXC + (int)c4[3];
    map[(size_t)b * NCELL + idx] = p;
}

// ---- Pass 2: gather + stream canvas --------------------------------------
// Block = 64 cells x 64 channels of one batch. 256 threads:
//   cellLocal = tid & 63 (lane-fast -> coalesced stores along nx/ny)
//   cGroup    = tid >> 6 handles channels [cGroup*16, cGroup*16+16)
// Map tile is copied global->LDS with the CDNA5 async DMA path, then each
// thread does 4x float4 feature gathers and 16x NT canvas stores.
__global__ __launch_bounds__(256) void pps_gather(
    const float* __restrict__ feat, const int* __restrict__ map,
    float* __restrict__ out, int B, int P) {
    __shared__ int s_pillar[64];
    const int tilesPerBatch = NCELL / 64; // 3348
    int tile = blockIdx.x;
    int b = tile / tilesPerBatch;
    int cell0 = (tile - b * tilesPerBatch) * 64;

    int tid = threadIdx.x;
    int cellLocal = tid & 63;
    int cGroup = tid >> 6; // 0..3

    // Async global->LDS copy of the 64-entry map tile (waves 0-1 only).
    // LDS byte address = low 32 bits of the flat shared-memory address.
    const int* gbase = map + (size_t)b * NCELL + cell0; // block-uniform -> SGPRs
    if (tid < 64) {
        unsigned lds_addr = (unsigned)(uintptr_t)(&s_pillar[tid]);
        unsigned voff = (unsigned)tid * 4u;
        asm volatile("global_load_async_to_lds_b32 %0, %1, %2"
                     :
                     : "v"(lds_addr), "v"(voff),
                       "s"((unsigned long long)(uintptr_t)gbase)
                     : "memory");
        asm volatile("s_wait_asynccnt 0x0" ::: "memory");
    }
    __syncthreads();

    int pil = s_pillar[cellLocal];
    const float* fbase = feat + ((size_t)b * P + (pil >= 0 ? pil : 0)) * 64;
    if (pil >= 0) __builtin_prefetch(fbase + cGroup * 16, 0, 0); // global_prefetch
    float* obase = out + (size_t)b * 64 * NCELL + cell0 + cellLocal;

#pragma unroll
    for (int k = 0; k < 4; ++k) {
        int c = cGroup * 16 + k * 4;
        float4 v = make_float4(0.f, 0.f, 0.f, 0.f);
        if (pil >= 0) v = *(const float4*)(fbase + c); // global_load_b128 gather
        // Write-once 439MB stream (> 192MB L2): nontemporal so the gather
        // working set (feat 41MB + map 7MB) stays resident in L2.
        __builtin_nontemporal_store(v.x, obase + (size_t)(c + 0) * NCELL);
        __builtin_nontemporal_store(v.y, obase + (size_t)(c + 1) * NCELL);
        __builtin_nontemporal_store(v.z, obase + (size_t)(c + 2) * NCELL);
        __builtin_nontemporal_store(v.w, obase + (size_t)(c + 3) * NCELL);
    }
}

// ---- Fallback path (generic C, or workspace too small): zero + scatter ---
__global__ void pps_zero(float* __restrict__ out, size_t n) {
    size_t i = (size_t)blockIdx.x * blockDim.x + threadIdx.x;
    size_t stride = (size_t)gridDim.x * blockDim.x;
    for (; i < n; i += stride) __builtin_nontemporal_store(0.f, out + i);
}

__global__ void pps_scatter(const float* __restrict__ feat,
                            const long long* __restrict__ coords,
                            float* __restrict__ out, int B, int P, int C) {
    size_t i = (size_t)blockIdx.x * blockDim.x + threadIdx.x;
    size_t total = (size_t)B * P * C;
    if (i >= total) return;
    int c = (int)(i % C);
    size_t bp = i / C;
    int b = (int)(bp / P);
    const long long* c4 = coords + bp * 4;
    int idx = (int)c4[2] * NXC + (int)c4[3];
    out[((size_t)b * C + c) * NCELL + idx] = feat[i];
}

extern "C" void kernel_launch(void* const* d_in, const int* in_sizes, int n_in,
                              void* d_out, int out_size, void* d_ws, size_t ws_size,
                              hipStream_t stream) {
    const float* feat = (const float*)d_in[0];
    const long long* coords = (const long long*)d_in[1]; // int64 coords
    float* out = (float*)d_out;

    // Derive shapes: in_sizes[0] = B*P*C, in_sizes[1] = B*P*4,
    // out_size = B*C*NCELL.
    int BP = in_sizes[1] / 4;
    int C = in_sizes[0] / BP;
    int B = out_size / (C * NCELL);
    int P = BP / B;

    size_t ws_need = (size_t)B * NCELL * sizeof(int);

    if (C == 64 && ws_size >= ws_need) {
        int* map = (int*)d_ws;
        int nmap = B * NCELL;
        pps_init_map<<<(nmap + 255) / 256, 256, 0, stream>>>(map, nmap);
        pps_build_map<<<(B * P + 255) / 256, 256, 0, stream>>>(coords, map, B, P);
        int nblocks = B * (NCELL / 64);
        pps_gather<<<nblocks, 256, 0, stream>>>(feat, map, out, B, P);
    } else {
        size_t n = (size_t)out_size;
        int zb = (int)((n + 4 * 256 - 1) / (4 * 256));
        if (zb > 65535 * 16) zb = 65535 * 16;
        pps_zero<<<zb, 256, 0, stream>>>(out, n);
        size_t total = (size_t)B * P * C;
        pps_scatter<<<(int)((total + 255) / 256), 256, 0, stream>>>(feat, coords, out, B, P, C);
    }
}